// InnerComp_34076270526661
// MI455X (gfx1250) — compile-verified
//
#include <hip/hip_runtime.h>
#include <hip/hip_bf16.h>

#define BS 256
#define LL 512
#define DD 64
#define THRESH (1.0f / 256.0f)

typedef __attribute__((ext_vector_type(2))) float v2f;
typedef __attribute__((ext_vector_type(4))) float v4f;
typedef __attribute__((ext_vector_type(8))) float v8f;

// ---------------------------------------------------------------------------
// Kernel 1: per-batch Gram max  m[b] = max_{l,m} <seq[b,l], seq[b,m]>
// One block (8 waves) per batch. seq[b] (512x64 f32 = 128KB) staged to LDS via
// async global->LDS copies. i-major tile loop: A fragment hoisted per row-tile,
// B streamed over j = i..31. Wave w owns rows {w, 31-w, 8+w, 23-w} -> exactly
// 66 tile pairs per wave (perfect balance). All loop control is scalar
// (readfirstlane) so EXEC stays all-ones around every WMMA.
// ---------------------------------------------------------------------------
__global__ __launch_bounds__(256) void gram_max_kernel(
    const float* __restrict__ seq, float* __restrict__ mmax) {
  __shared__ float smem[LL * DD];   // 128 KB (of 320 KB/WGP)
  __shared__ float red[256];

  const int b    = blockIdx.x;
  const int tid  = threadIdx.x;
  const int lane = tid & 31;
  const int wv   = __builtin_amdgcn_readfirstlane(tid >> 5);  // scalar wave id

  // --- async stage: 256 threads x 32 x b128 = 131072 bytes ---
  {
    const float* gbase = seq + (size_t)b * (LL * DD);
    unsigned ldsbase = (unsigned)(unsigned long long)(&smem[0]);
#pragma unroll
    for (int c = 0; c < 32; ++c) {
      unsigned byteoff = (unsigned)((c * 256 + tid) * 16);
      unsigned ldsaddr = ldsbase + byteoff;
      asm volatile("global_load_async_to_lds_b128 %0, %1, %2"
                   :
                   : "v"(ldsaddr), "v"(byteoff), "s"(gbase)
                   : "memory");
    }
    asm volatile("s_wait_asynccnt 0" ::: "memory");
  }
  __syncthreads();

  // --- i-major tile loop ---
  const int ro = lane & 15;        // row within tile
  const int ko = 2 * (lane >> 4);  // K sub-pair select {0,1} vs {2,3}
  float mloc = -3.402823466e38f;

  const int rows[4] = {wv, 31 - wv, 8 + wv, 23 - wv};

#pragma unroll 1
  for (int r = 0; r < 4; ++r) {
    const int i = rows[r];
    const int ra = i * 16 + ro;

    v2f a[16];
#pragma unroll
    for (int kk = 0; kk < 16; ++kk)
      a[kk] = *(const v2f*)&smem[ra * DD + kk * 4 + ko];

#pragma unroll 1
    for (int j = i; j < 32; ++j) {
      const int rb = j * 16 + ro;
      v2f bv[16];
#pragma unroll
      for (int kk = 0; kk < 16; ++kk)
        bv[kk] = *(const v2f*)&smem[rb * DD + kk * 4 + ko];

      v8f acc = {0.f, 0.f, 0.f, 0.f, 0.f, 0.f, 0.f, 0.f};
#pragma unroll
      for (int kk = 0; kk < 16; ++kk) {
        acc = __builtin_amdgcn_wmma_f32_16x16x4_f32(
            false, a[kk], false, bv[kk], (short)0, acc, false, false);
      }
#pragma unroll
      for (int e = 0; e < 8; ++e) mloc = fmaxf(mloc, acc[e]);
    }
  }

  red[tid] = mloc;
  __syncthreads();
  for (int s = 128; s > 0; s >>= 1) {
    if (tid < s) red[tid] = fmaxf(red[tid], red[tid + s]);
    __syncthreads();
  }
  if (tid == 0) mmax[b] = red[0];
}

// ---------------------------------------------------------------------------
// Kernel 2: softmax over m[0..255], threshold mask, w[j] = wb[j]*mask[j],
// and sum(wb). Single block of 256 threads.
// ---------------------------------------------------------------------------
__global__ __launch_bounds__(256) void softmax_mask_kernel(
    const float* __restrict__ mmax, const float* __restrict__ wb,
    float* __restrict__ w_out, float* __restrict__ sumwb_out) {
  __shared__ float red[256];
  const int tid = threadIdx.x;

  float mv = mmax[tid];
  red[tid] = mv;
  __syncthreads();
  for (int s = 128; s > 0; s >>= 1) {
    if (tid < s) red[tid] = fmaxf(red[tid], red[tid + s]);
    __syncthreads();
  }
  const float M = red[0];
  __syncthreads();

  const float e = expf(mv - M);
  red[tid] = e;
  __syncthreads();
  for (int s = 128; s > 0; s >>= 1) {
    if (tid < s) red[tid] += red[tid + s];
    __syncthreads();
  }
  const float denom = red[0];
  __syncthreads();

  const float gb  = e / denom;
  const float wbv = wb[tid];
  w_out[tid] = (gb > THRESH) ? wbv : 0.0f;

  red[tid] = wbv;
  __syncthreads();
  for (int s = 128; s > 0; s >>= 1) {
    if (tid < s) red[tid] += red[tid + s];
    __syncthreads();
  }
  if (tid == 0) sumwb_out[0] = red[0];
}

// ---------------------------------------------------------------------------
// Kernel 3a (fused): S[l,d] = sum_j w[j]*seq[j,l,d]  AND  out[j, 0:L, :] =
// seq[j] (the copy rides the same streaming read of seq -> saves 33.5 MB of
// HBM traffic). 32 blocks x 256 threads; one float4 of [512,64] per thread.
// ---------------------------------------------------------------------------
__global__ __launch_bounds__(256) void wsum_copy_kernel(
    const float* __restrict__ seq, const float* __restrict__ w,
    float* __restrict__ S, float* __restrict__ out) {
  __shared__ float wl[256];
  const int tid = threadIdx.x;
  wl[tid] = w[tid];
  __syncthreads();

  const int idx4 = blockIdx.x * 256 + tid;  // 0 .. 8191 (float4 units of L*D)
  const v4f* sp = (const v4f*)seq;
  v4f* op = (v4f*)out;
  v4f acc = {0.f, 0.f, 0.f, 0.f};
#pragma unroll 4
  for (int j = 0; j < BS; ++j) {
    v4f x = sp[(size_t)j * (LL * DD / 4) + idx4];
    acc = acc + x * wl[j];
    op[(size_t)j * (2 * LL * DD / 4) + idx4] = x;  // out[j, 0:L, :] = seq[j]
  }
  ((v4f*)S)[idx4] = acc;
}

// ---------------------------------------------------------------------------
// Kernel 3b: Y = S @ Wn^T + sum(wb)*bn + bb  via f32 WMMA.
// 128 output tiles (32 row-tiles x 4 col-tiles), 4 blocks x 8 waves x 4 tiles.
// ---------------------------------------------------------------------------
__global__ __launch_bounds__(256) void y_gemm_kernel(
    const float* __restrict__ S, const float* __restrict__ Wn,
    const float* __restrict__ bn, const float* __restrict__ bb,
    const float* __restrict__ sumwb, float* __restrict__ Y) {
  const int tid  = threadIdx.x;
  const int lane = tid & 31;
  const int wv   = __builtin_amdgcn_readfirstlane(tid >> 5);
  const int gid  = blockIdx.x * 8 + wv;  // 0..31

  const float sw  = sumwb[0];
  const float bbv = bb[0];
  const int ro = lane & 15;
  const int ko = 2 * (lane >> 4);

#pragma unroll 1
  for (int tt = 0; tt < 4; ++tt) {
    const int t = gid * 4 + tt;          // 0..127
    const int rowbase = (t >> 2) * 16;   // over L=512
    const int colbase = (t & 3) * 16;    // over D=64
    const int ra = rowbase + ro;         // A row in S
    const int rb = colbase + ro;         // B "row" = Wn row (B = Wn^T column)

    v2f a[16], bv[16];
#pragma unroll
    for (int kk = 0; kk < 16; ++kk) {
      a[kk]  = *(const v2f*)&S [ra * DD + kk * 4 + ko];
      bv[kk] = *(const v2f*)&Wn[rb * DD + kk * 4 + ko];
    }

    v8f acc = {0.f, 0.f, 0.f, 0.f, 0.f, 0.f, 0.f, 0.f};
#pragma unroll
    for (int kk = 0; kk < 16; ++kk) {
      acc = __builtin_amdgcn_wmma_f32_16x16x4_f32(
          false, a[kk], false, bv[kk], (short)0, acc, false, false);
    }

    // D layout: lane<16 -> M=v, N=lane; lane>=16 -> M=v+8, N=lane-16
    const int N     = colbase + (lane & 15);
    const int Mbase = rowbase + ((lane >> 4) ? 8 : 0);
    const float cadd = sw * bn[N] + bbv;
#pragma unroll
    for (int v = 0; v < 8; ++v) {
      Y[(Mbase + v) * DD + N] = acc[v] + cadd;
    }
  }
}

// ---------------------------------------------------------------------------
// Kernel 3c: out[b, L:2L, :] = Y  (broadcast; Y is 128KB -> L2 resident).
// grid = (32, 256); float4 per thread.
// ---------------------------------------------------------------------------
__global__ __launch_bounds__(256) void bcast_y_kernel(
    const float* __restrict__ Y, float* __restrict__ out) {
  const int b  = blockIdx.y;
  const int i4 = blockIdx.x * 256 + threadIdx.x;  // 0..8191 (float4 units)
  ((v4f*)out)[(size_t)b * (2 * LL * DD / 4) + (LL * DD / 4) + i4] =
      ((const v4f*)Y)[i4];
}

// ---------------------------------------------------------------------------
extern "C" void kernel_launch(void* const* d_in, const int* in_sizes, int n_in,
                              void* d_out, int out_size, void* d_ws,
                              size_t ws_size, hipStream_t stream) {
  const float* seq = (const float*)d_in[0];  // [256,512,64]
  const float* Wn  = (const float*)d_in[1];  // [64,64]
  const float* bn  = (const float*)d_in[2];  // [64]
  const float* wb  = (const float*)d_in[3];  // [256]
  const float* bb  = (const float*)d_in[4];  // [1]
  float* out = (float*)d_out;                // [256,1024,64]

  char* ws = (char*)d_ws;
  float* mmax  = (float*)(ws + 0);                    // 256 f32
  float* w     = (float*)(ws + 1024);                 // 256 f32
  float* sumwb = (float*)(ws + 2048);                 // 1 f32
  float* S     = (float*)(ws + 4096);                 // 512*64 f32
  float* Y     = (float*)(ws + 4096 + LL * DD * 4);   // 512*64 f32

  gram_max_kernel<<<BS, 256, 0, stream>>>(seq, mmax);
  softmax_mask_kernel<<<1, 256, 0, stream>>>(mmax, wb, w, sumwb);
  wsum_copy_kernel<<<LL * DD / (256 * 4), 256, 0, stream>>>(seq, w, S, out);
  y_gemm_kernel<<<4, 256, 0, stream>>>(S, Wn, bn, bb, sumwb, Y);
  bcast_y_kernel<<<dim3(LL * DD / (256 * 4), BS), 256, 0, stream>>>(Y, out);
}